// ModifiedSTNGPe_66915590472098
// MI455X (gfx1250) — compile-verified
//
#include <hip/hip_runtime.h>

typedef __attribute__((ext_vector_type(2))) float v2f;
typedef __attribute__((ext_vector_type(8))) float v8f;

#define U       20      // UNITS
#define G       80      // 4*U gate columns
#define TILE_M  16
#define WAVES   8       // waves per block

__device__ __forceinline__ float sigm(float v) {
    // robust at extremes: exp(+inf)->inf -> 0 ; exp(-big)->0 -> 1
    return 1.0f / (1.0f + __expf(-v));
}
__device__ __forceinline__ float tanh_fast(float v) {
    // 1 - 2/(exp(2v)+1): saturates exactly to +/-1, no inf/inf NaN
    return 1.0f - 2.0f / (__expf(2.0f * v) + 1.0f);
}

__global__ __launch_bounds__(WAVES * 32)
void ModifiedSTNGPe_lstm_wmma(const float* __restrict__ x,
                              const float* __restrict__ h,
                              const float* __restrict__ c,
                              const float* __restrict__ W_ih,
                              const float* __restrict__ W_hh,
                              const float* __restrict__ b_ih,
                              const float* __restrict__ b_hh,
                              const float* __restrict__ W_lat,
                              float* __restrict__ h_out,
                              float* __restrict__ c_out,
                              int nTiles, int Btot)
{
    __shared__ float s_g [WAVES * TILE_M * G];  // per-wave 16x80 gate tile (40 KB)
    __shared__ float s_hl[WAVES * TILE_M * U];  // per-wave 16x20 h_lstm tile (10 KB)

    const int lane = threadIdx.x & 31;
    const int wib  = threadIdx.x >> 5;
    const int wave = blockIdx.x * WAVES + wib;
    if (wave >= nTiles) return;            // wave-uniform: EXEC stays all-1s

    float* gw = s_g  + wib * (TILE_M * G);
    float* hw = s_hl + wib * (TILE_M * U);

    const int r0   = wave * TILE_M;
    const int m    = lane & 15;            // row within tile (A/B lane row/col)
    const int half = lane >> 4;            // K-half select for f32 WMMA frags

    // Clamp row for loads near a partial tail tile (stores are guarded below).
    const int rowm = min(r0 + m, Btot - 1);
    const float* xrow = x + (size_t)rowm * U;
    const float* hrow = h + (size_t)rowm * U;

    // ---- init gate accumulators with broadcast bias (bias depends only on N) ----
    v8f acc[5];
#pragma unroll
    for (int nt = 0; nt < 5; ++nt) {
        float bv = b_ih[nt * 16 + m] + b_hh[nt * 16 + m];
        v8f t;
#pragma unroll
        for (int v = 0; v < 8; ++v) t[v] = bv;
        acc[nt] = t;
    }

    // ---- gates = [x|h] @ [W_ih|W_hh]^T + b : 5 N-tiles x 10 K-steps of f32 WMMA ----
#pragma unroll
    for (int s = 0; s < 10; ++s) {
        const int k = 4 * s + 2 * half;            // even; pair (k,k+1) never straddles 20
        v2f a = (k < U) ? *(const v2f*)(xrow + k)
                        : *(const v2f*)(hrow + (k - U));
#pragma unroll
        for (int nt = 0; nt < 5; ++nt) {
            const int n = nt * 16 + m;             // gate column 0..79
            v2f b = (k < U) ? *(const v2f*)(W_ih + n * U + k)
                            : *(const v2f*)(W_hh + n * U + (k - U));
            acc[nt] = __builtin_amdgcn_wmma_f32_16x16x4_f32(
                          false, a, false, b, (short)0, acc[nt], false, false);
        }
    }

    // ---- spill gate tiles to this wave's LDS slice ----
#pragma unroll
    for (int nt = 0; nt < 5; ++nt)
#pragma unroll
        for (int v = 0; v < 8; ++v) {
            const int mm = v + 8 * half;           // C layout: M = v (+8 for hi lanes)
            gw[mm * G + nt * 16 + m] = acc[nt][v];
        }

    // ---- elementwise LSTM: coalesced c load / c_next store, h_lstm -> LDS ----
    const int nvalid = (Btot - r0) * U;            // valid flat elems in this tile
    const float* crow  = c     + (size_t)r0 * U;
    float*       cnrow = c_out + (size_t)r0 * U;
#pragma unroll
    for (int j = 0; j < 10; ++j) {
        const int e  = j * 32 + lane;              // e in [0,320), row-major (m,u)
        const int mm = e / U;
        const int u  = e - mm * U;
        const float gi = gw[mm * G +          u];
        const float gf = gw[mm * G +     U  + u];
        const float gg = gw[mm * G + 2 * U  + u];
        const float go = gw[mm * G + 3 * U  + u];
        const float cv = crow[min(e, nvalid - 1)];
        const float cn = sigm(gf) * cv + sigm(gi) * tanh_fast(gg);
        if (e < nvalid) cnrow[e] = cn;
        hw[e] = sigm(go) * tanh_fast(cn);          // h_lstm, flat layout == m*U+u
    }

    // ---- lateral: h_lstm @ W_lat^T via WMMA (N padded 20->32, pad cols read 0) ----
    v8f lacc[2];
#pragma unroll
    for (int nt = 0; nt < 2; ++nt) {
        v8f t;
#pragma unroll
        for (int v = 0; v < 8; ++v) t[v] = 0.0f;
        lacc[nt] = t;
    }
#pragma unroll
    for (int s = 0; s < 5; ++s) {
        const int k = 4 * s + 2 * half;            // 0..18
        v2f a;
        a.x = hw[m * U + k];
        a.y = hw[m * U + k + 1];
#pragma unroll
        for (int nt = 0; nt < 2; ++nt) {
            const int n = nt * 16 + m;
            v2f b;
            if (n < U) b = *(const v2f*)(W_lat + n * U + k);
            else       { b.x = 0.0f; b.y = 0.0f; }
            lacc[nt] = __builtin_amdgcn_wmma_f32_16x16x4_f32(
                           false, a, false, b, (short)0, lacc[nt], false, false);
        }
    }

    // ---- spill lateral result (reuse gate LDS slice; intra-wave LDS is in-order) ----
#pragma unroll
    for (int nt = 0; nt < 2; ++nt)
#pragma unroll
        for (int v = 0; v < 8; ++v) {
            const int n = nt * 16 + m;
            if (n < U) {
                const int mm = v + 8 * half;
                gw[mm * U + n] = lacc[nt][v];
            }
        }

    // ---- h_next = tanh(lateral), coalesced store ----
    float* hnrow = h_out + (size_t)r0 * U;
#pragma unroll
    for (int j = 0; j < 10; ++j) {
        const int e = j * 32 + lane;
        if (e < nvalid) hnrow[e] = tanh_fast(gw[e]);
    }
}

extern "C" void kernel_launch(void* const* d_in, const int* in_sizes, int n_in,
                              void* d_out, int out_size, void* d_ws, size_t ws_size,
                              hipStream_t stream) {
    const float* x     = (const float*)d_in[0];
    const float* h     = (const float*)d_in[1];
    const float* c     = (const float*)d_in[2];
    const float* W_ih  = (const float*)d_in[3];
    const float* W_hh  = (const float*)d_in[4];
    const float* b_ih  = (const float*)d_in[5];
    const float* b_hh  = (const float*)d_in[6];
    const float* W_lat = (const float*)d_in[7];

    const int B      = in_sizes[0] / U;                    // 524288
    const int nTiles = (B + TILE_M - 1) / TILE_M;          // 32768
    float* h_out = (float*)d_out;
    float* c_out = (float*)d_out + (size_t)B * U;

    const int blocks = (nTiles + WAVES - 1) / WAVES;       // 4096
    ModifiedSTNGPe_lstm_wmma<<<blocks, WAVES * 32, 0, stream>>>(
        x, h, c, W_ih, W_hh, b_ih, b_hh, W_lat, h_out, c_out, nTiles, B);
}